// SegmentTreeAttention_2465311228074
// MI455X (gfx1250) — compile-verified
//
#include <hip/hip_runtime.h>

// ---------------- problem constants (from reference) ----------------
#define BB      16
#define SS      8192
#define DD      64
#define NN      32
#define NLEVELS 5

// ---------------- tiling ----------------
#define WAVES   8
#define QPW     16              // queries per wave = WMMA M
#define QPB     (WAVES * QPW)   // 128 queries per block
#define PSTR    66              // padded row stride for Pk/Pv (breaks bank conflicts)
#define SSTR    33              // padded stride for per-wave S (dot) tile
#define CSTR    34              // padded stride for per-wave coefficient tile (even for b64)

typedef __attribute__((ext_vector_type(8))) float v8f;
typedef __attribute__((ext_vector_type(2))) float v2f;

__device__ __forceinline__ v8f wmma_f32_k4(v2f a, v2f b, v8f c) {
  // D = A(16x4 f32) * B(4x16 f32) + C(16x16 f32), full fp32 matrix op
  return __builtin_amdgcn_wmma_f32_16x16x4_f32(false, a, false, b, (short)0, c,
                                               false, false);
}

__device__ __forceinline__ float sigmoidf_fast(float x) {
  return 1.0f / (1.0f + __expf(-x));
}

__global__ __launch_bounds__(256)
void segtree_attn_wmma(const float* __restrict__ Q,
                       const float* __restrict__ K,
                       const float* __restrict__ V,
                       const int*   __restrict__ vlen,
                       float* __restrict__ Out) {
  __shared__ float Pk[NN][PSTR];              // key prefix sums   (8.4 KB)
  __shared__ float Pv[NN][PSTR];              // value prefix sums (8.4 KB)
  __shared__ float Sld[WAVES][QPW][SSTR];     // per-wave dot tile S[q][m]
  __shared__ float Cld[WAVES][QPW][CSTR];     // per-wave coefficient tile C[q][m]

  const int b    = blockIdx.x;                // batch
  const int t    = threadIdx.x;
  const int wave = t >> 5;                    // wave32
  const int lane = t & 31;

  const float* Qblk = Q + ((size_t)b * SS + (size_t)blockIdx.y * QPB) * DD;

  // Prefetch this block's entire 32 KB query tile (256 thr x 128 B lines)
  __builtin_prefetch(Qblk + (size_t)t * 32, 0, 0);

  // ---- Phase 1: prefix sums of keys/values into LDS (P[0]=0, P[m]=sum x[1..m]) ----
  if (t < 2 * DD) {
    const int d = t & (DD - 1);
    const float* src = (t < DD) ? (K + (size_t)b * NN * DD)
                                : (V + (size_t)b * NN * DD);
    float (*P)[PSTR] = (t < DD) ? Pk : Pv;
    float acc = 0.0f;
    P[0][d] = 0.0f;
    #pragma unroll
    for (int m = 1; m < NN; ++m) {
      acc += src[m * DD + d];
      P[m][d] = acc;
    }
  }
  // zero this wave's coefficient tile
  {
    float* c = &Cld[wave][0][0];
    for (int i = lane; i < QPW * CSTR; i += 32) c[i] = 0.0f;
  }
  __syncthreads();

  // fragment coordinates shared by both GEMMs (16x16x4 f32 layout):
  // lanes 0-15 hold K=0,1 ; lanes 16-31 hold K=2,3
  const int row  = lane & 15;
  const int koff = (lane >> 4) << 1;

  const float* Qt = Qblk + (size_t)wave * QPW * DD;

  // ---- Phase 2: GEMM1  S[16q x 32m] = Q_tile[16x64] * Pk^T[64x32] ----
  v8f accA = {}; // m = 0..15
  v8f accB = {}; // m = 16..31
  #pragma unroll
  for (int k0 = 0; k0 < DD; k0 += 4) {
    const int kk = k0 + koff;
    v2f a = *(const v2f*)(Qt + row * DD + kk);          // A[row][kk..kk+1]
    v2f b0; b0.x = Pk[row][kk];      b0.y = Pk[row][kk + 1];       // B[k][m=row]
    v2f b1; b1.x = Pk[row + 16][kk]; b1.y = Pk[row + 16][kk + 1];  // B[k][m=row+16]
    accA = wmma_f32_k4(a, b0, accA);
    accB = wmma_f32_k4(a, b1, accB);
  }

  // scatter S fragments to LDS so each lane can walk its own query's 32 dots
  {
    const int mrow = (lane >> 4) << 3;   // 0 or 8
    const int nloc = lane & 15;
    #pragma unroll
    for (int j = 0; j < 8; ++j) {
      Sld[wave][mrow + j][nloc]      = accA[j];
      Sld[wave][mrow + j][nloc + 16] = accB[j];
    }
  }

  // ---- Phase 3: scalar segment-tree descent, one query per lane (lanes 0-15) ----
  if (lane < 16) {
    const float* Sq = Sld[wave][lane];
    float*       Cq = Cld[wave][lane];
    const int nv = vlen[b];
    int l = 1, r = NN;
    float pw = 0.0625f;                         // 0.5^(NLEVELS-1-i) weights
    #pragma unroll
    for (int lev = 0; lev < NLEVELS; ++lev) {
      const int mid = (l + r) >> 1;
      int  hiL  = min(mid, nv - 1);
      bool okL  = (hiL >= l);
      int  hicL = max(0, min(hiL, NN - 1));
      int  hiR  = min(r, nv - 1);
      bool okR  = (hiR >= mid + 1);
      int  hicR = max(0, min(hiR, NN - 1));
      const float dL = okL ? (Sq[hicL] - Sq[l - 1]) : 0.0f;
      const float dR = okR ? (Sq[hicR] - Sq[mid])   : 0.0f;
      const float ls = sigmoidf_fast(dL);
      const float rs = sigmoidf_fast(dR);
      const bool cond = (ls >= rs);
      if (cond) {
        if (okR) { Cq[hicR] += rs * pw; Cq[mid] -= rs * pw; }
        r = mid;
      } else {
        if (okL) { Cq[hicL] += ls * pw; Cq[l - 1] -= ls * pw; }
        l = mid + 1;
      }
      pw *= 2.0f;
    }
  }
  // same-wave DS ops are in order: GEMM2's LDS reads below see descent's writes

  // ---- Phase 4: GEMM2  ans[16q x 64d] = C[16x32] * Pv[32x64] ----
  v8f o0 = {}, o1 = {}, o2 = {}, o3 = {};
  #pragma unroll
  for (int k0 = 0; k0 < NN; k0 += 4) {
    const int kk = k0 + koff;
    v2f a2; a2.x = Cld[wave][row][kk]; a2.y = Cld[wave][row][kk + 1];
    v2f b0; b0.x = Pv[kk][row];      b0.y = Pv[kk + 1][row];
    v2f b1; b1.x = Pv[kk][row + 16]; b1.y = Pv[kk + 1][row + 16];
    v2f b2; b2.x = Pv[kk][row + 32]; b2.y = Pv[kk + 1][row + 32];
    v2f b3; b3.x = Pv[kk][row + 48]; b3.y = Pv[kk + 1][row + 48];
    o0 = wmma_f32_k4(a2, b0, o0);
    o1 = wmma_f32_k4(a2, b1, o1);
    o2 = wmma_f32_k4(a2, b2, o2);
    o3 = wmma_f32_k4(a2, b3, o3);
  }

  // ---- Phase 5: store D fragments: VGPR j -> row (j | j+8), lane -> column ----
  {
    float* Ot = Out + ((size_t)b * SS + (size_t)blockIdx.y * QPB +
                       (size_t)wave * QPW) * DD;
    const int mrow = (lane >> 4) << 3;
    const int dcol = lane & 15;
    #pragma unroll
    for (int j = 0; j < 8; ++j) {
      float* orow = Ot + (mrow + j) * DD + dcol;
      orow[0]  = o0[j];
      orow[16] = o1[j];
      orow[32] = o2[j];
      orow[48] = o3[j];
    }
  }
}

extern "C" void kernel_launch(void* const* d_in, const int* in_sizes, int n_in,
                              void* d_out, int out_size, void* d_ws, size_t ws_size,
                              hipStream_t stream) {
  const float* Q  = (const float*)d_in[0];   // [16, 8192, 64] f32
  const float* K  = (const float*)d_in[1];   // [16, 32, 64]   f32
  const float* V  = (const float*)d_in[2];   // [16, 32, 64]   f32
  const int*   vl = (const int*)d_in[3];     // [16]           int
  float* Out = (float*)d_out;                // [16, 8192, 64] f32

  dim3 grid(BB, SS / QPB);                   // (16, 64)
  segtree_attn_wmma<<<grid, 256, 0, stream>>>(Q, K, V, vl, Out);
}